// SAGE_46961172414795
// MI455X (gfx1250) — compile-verified
//
#include <hip/hip_runtime.h>

// ---------------- problem constants (match reference) ----------------
#define NN 50000        // nodes (= 3125 * 16, exact WMMA M tiling)
#define NE 800000       // edges
#define F0 128          // input feats
#define HID 256         // hidden
#define NCLS 64         // classes

typedef __attribute__((ext_vector_type(2))) float v2f;
typedef __attribute__((ext_vector_type(8))) float v8f;

// ---------------- utility kernels ----------------
__global__ void zero_f32(float* __restrict__ p, int n) {
    int i = blockIdx.x * blockDim.x + threadIdx.x;
    if (i < n) p[i] = 0.0f;
}

__global__ void degree_count(const int* __restrict__ dst, float* __restrict__ deg, int e) {
    int i = blockIdx.x * blockDim.x + threadIdx.x;
    if (i < e) atomicAdd(deg + dst[i], 1.0f);
}

__global__ void make_invdeg(float* __restrict__ deg, int n) {
    int i = blockIdx.x * blockDim.x + threadIdx.x;
    if (i < n) deg[i] = 1.0f / fmaxf(deg[i], 1.0f);
}

// Edge-parallel mean-aggregation scatter: coalesced float4 gather from the
// L2-resident feature matrix, f32 L2 atomics on scatter.
__global__ void scatter_edges(const float* __restrict__ H,
                              const int* __restrict__ src,
                              const int* __restrict__ dst,
                              float* __restrict__ Agg,
                              int e, int d, int cshift) {
    unsigned tid = blockIdx.x * blockDim.x + threadIdx.x;
    int edge = (int)(tid >> cshift);
    int c = (int)((tid & ((1u << cshift) - 1u)) << 2);
    if (edge < e) {
        int s = src[edge];
        int t = dst[edge];
        const float4 v = *(const float4*)(H + (size_t)s * d + c);
        float* p = Agg + (size_t)t * d + c;
        atomicAdd(p + 0, v.x);
        atomicAdd(p + 1, v.y);
        atomicAdd(p + 2, v.z);
        atomicAdd(p + 3, v.w);
    }
}

// Pack BOTH weight matrices of a layer into one fragment-ordered stream:
// float index = (((w*KC + c)*4 + t)*2 + m)*64 + L*2 + j
//   w = owning wave (n>>6), t = n-tile within wave ((n>>4)&3), m = 0:Ws 1:Wn
//   c = k>>2, lane L = (((k>>1)&1)<<4) | (n&15), j = k&1
// => per (wave, K-chunk): 8 fragments x 256B contiguous (2KB block), so the
// GEMM inner loop is ONE pointer + immediate offsets, clause-friendly.
__global__ void pack_pair(const float* __restrict__ Ws, const float* __restrict__ Wn,
                          float* __restrict__ Wp, int K, int N) {
    int idx = blockIdx.x * blockDim.x + threadIdx.x;
    int kn = K * N;
    if (idx < 2 * kn) {
        int m = idx >= kn ? 1 : 0;
        int rem = idx - m * kn;
        int k = rem / N;
        int n = rem - k * N;
        int tg = n >> 4, w = tg >> 2, t = tg & 3, r = n & 15;
        int c = k >> 2, half = (k >> 1) & 1, j = k & 1;
        int L = (half << 4) | r;
        const float* src = m ? Wn : Ws;
        Wp[((((size_t)w * (K >> 2) + c) * 4 + t) * 2 + m) * 64 + L * 2 + j] = src[rem];
    }
}

// Single-matrix pack for the FC layer (N=64 -> 4 tiles, one shared slab):
// float index = ((c*4 + t)*64) + L*2 + j  (128 v2f per K-chunk)
__global__ void pack_single(const float* __restrict__ W, float* __restrict__ Wp,
                            int K, int N) {
    int idx = blockIdx.x * blockDim.x + threadIdx.x;
    if (idx < K * N) {
        int k = idx / N;
        int n = idx - k * N;
        int t = n >> 4, r = n & 15;
        int c = k >> 2, half = (k >> 1) & 1, j = k & 1;
        int L = (half << 4) | r;
        Wp[((size_t)c * 4 + t) * 64 + L * 2 + j] = W[idx];
    }
}

// ---------------- fused SAGE layer GEMM core ----------------
// acc = Hs @ Ws + (Agg*invdeg) @ Wn for MT M-tiles x 4 N-tiles per wave.
// Per K-chunk: 2*MT A-loads (b64) + 8 B-loads (b64, one clause) -> 8*MT WMMAs.
template <int MT>
__device__ __forceinline__ void sage_strip(const float* __restrict__ Hs,
                                           const float* __restrict__ Agg,
                                           const float* __restrict__ invdeg,
                                           const v2f* __restrict__ pw,
                                           const float* __restrict__ bias,
                                           float* __restrict__ Hout,
                                           int K, int N, int row0,
                                           int lane, int wave) {
    const int r = lane & 15;
    const int halfk = (lane >> 4) << 1;
    const int KC = K >> 2;

    float id[MT];
    const float* hrow[MT];
    const float* arow[MT];
#pragma unroll
    for (int m = 0; m < MT; ++m) {
        id[m] = invdeg[row0 + 16 * m + r];
        hrow[m] = Hs + (size_t)(row0 + 16 * m + r) * K + halfk;
        arow[m] = Agg + (size_t)(row0 + 16 * m + r) * K + halfk;
    }

    v8f acc[MT][4] = {};
#pragma unroll 2
    for (int c = 0; c < KC; ++c) {
        // all 8 B fragments for this chunk: contiguous 2KB, one clause
        v2f b[8];
#pragma unroll
        for (int q = 0; q < 8; ++q)
            b[q] = pw[(size_t)c * 256 + q * 32];
#pragma unroll
        for (int m = 0; m < MT; ++m) {
            v2f aS = *(const v2f*)(hrow[m] + 4 * c);
            v2f aN = *(const v2f*)(arow[m] + 4 * c);
            aN.x *= id[m];
            aN.y *= id[m];
#pragma unroll
            for (int t = 0; t < 4; ++t) {
                acc[m][t] = __builtin_amdgcn_wmma_f32_16x16x4_f32(
                    false, aS, false, b[2 * t + 0], (short)0, acc[m][t], false, false);
                acc[m][t] = __builtin_amdgcn_wmma_f32_16x16x4_f32(
                    false, aN, false, b[2 * t + 1], (short)0, acc[m][t], false, false);
            }
        }
    }

    // C/D layout: vgpr i -> row (i + 8*(lane>>4)), col = lane&15
    const int mboff = (lane >> 4) << 3;
#pragma unroll
    for (int m = 0; m < MT; ++m) {
        const int mb = row0 + 16 * m + mboff;
#pragma unroll
        for (int t = 0; t < 4; ++t) {
            const int n = ((4 * wave + t) << 4) + r;
            const float bn = bias[n];
#pragma unroll
            for (int i = 0; i < 8; ++i)
                Hout[(size_t)(mb + i) * N + n] = acc[m][t][i] + bn;
        }
    }
}

// Block = 128 threads (4 waves); wave w owns n-tiles 4w..4w+3 (N must be 256).
// Each block covers 2 M-tiles (32 rows); odd final tile uses the MT=1 path
// via a wave-uniform branch (EXEC stays all-ones around every WMMA).
__global__ void sage_gemm_wmma(const float* __restrict__ Hs,
                               const float* __restrict__ Agg,
                               const float* __restrict__ invdeg,
                               const float* __restrict__ Wpk,
                               const float* __restrict__ bias,
                               float* __restrict__ Hout,
                               int K, int N, int Mtiles) {
    const int lane = threadIdx.x & 31;
    const int wave = threadIdx.x >> 5;
    const int KC = K >> 2;
    const v2f* pw = (const v2f*)Wpk + (size_t)wave * KC * 256 + lane;
    const int row0 = blockIdx.y << 5;
    if ((int)(blockIdx.y * 2 + 1) < Mtiles)
        sage_strip<2>(Hs, Agg, invdeg, pw, bias, Hout, K, N, row0, lane, wave);
    else
        sage_strip<1>(Hs, Agg, invdeg, pw, bias, Hout, K, N, row0, lane, wave);
}

// Final FC: Out = H @ W + bias, N=64 (4 tiles per wave, packed single slab).
// Block = 128 (4 waves); each wave owns one M-tile; uniform early-out.
__global__ void fc_gemm_wmma(const float* __restrict__ H,
                             const float* __restrict__ WP,
                             const float* __restrict__ bias,
                             float* __restrict__ Out,
                             int K, int N, int Mtiles) {
    const int lane = threadIdx.x & 31;
    const int wave = threadIdx.x >> 5;
    const int mt = blockIdx.y * 4 + wave;
    if (mt >= Mtiles) return;                    // uniform per wave
    const int row0 = mt << 4;
    const int r = lane & 15;
    const int halfk = (lane >> 4) << 1;
    const int KC = K >> 2;

    const float* hrow = H + (size_t)(row0 + r) * K + halfk;
    const v2f* pw = (const v2f*)WP + lane;

    v8f acc[4] = {};
#pragma unroll 4
    for (int c = 0; c < KC; ++c) {
        v2f b[4];
#pragma unroll
        for (int t = 0; t < 4; ++t)
            b[t] = pw[(size_t)c * 128 + t * 32];
        v2f a = *(const v2f*)(hrow + 4 * c);
#pragma unroll
        for (int t = 0; t < 4; ++t)
            acc[t] = __builtin_amdgcn_wmma_f32_16x16x4_f32(
                false, a, false, b[t], (short)0, acc[t], false, false);
    }

    const int mb = row0 + ((lane >> 4) << 3);
#pragma unroll
    for (int t = 0; t < 4; ++t) {
        const int n = (t << 4) + r;
        const float bn = bias[n];
#pragma unroll
        for (int i = 0; i < 8; ++i)
            Out[(size_t)(mb + i) * N + n] = acc[t][i] + bn;
    }
}

// ---------------- host launcher ----------------
extern "C" void kernel_launch(void* const* d_in, const int* in_sizes, int n_in,
                              void* d_out, int out_size, void* d_ws, size_t ws_size,
                              hipStream_t stream) {
    const float* x    = (const float*)d_in[0];
    const int*   esrc = (const int*)d_in[1];
    const int*   edst = (const int*)d_in[2];
    const float* W0s  = (const float*)d_in[3];
    const float* W0n  = (const float*)d_in[4];
    const float* b0   = (const float*)d_in[5];
    const float* W1s  = (const float*)d_in[6];
    const float* W1n  = (const float*)d_in[7];
    const float* b1   = (const float*)d_in[8];
    const float* W2s  = (const float*)d_in[9];
    const float* W2n  = (const float*)d_in[10];
    const float* b2   = (const float*)d_in[11];
    const float* Wfc  = (const float*)d_in[12];
    const float* bfc  = (const float*)d_in[13];
    float* out = (float*)d_out;

    // workspace (floats): deg | agg | hA | hB | packed weights
    float* deg = (float*)d_ws;
    float* agg = deg + NN;
    float* hA  = agg + (size_t)NN * HID;
    float* hB  = hA + (size_t)NN * HID;
    float* p0  = hB + (size_t)NN * HID;              // 2*128*256
    float* p1  = p0 + (size_t)2 * F0 * HID;          // 2*256*256
    float* p2  = p1 + (size_t)2 * HID * HID;         // 2*256*256
    float* pfc = p2 + (size_t)2 * HID * HID;         // 256*64

    const int T = 256;
    const int MT = NN / 16;                          // 3125 M-tiles

    // one-time weight repacks (~1.4 MB total)
    pack_pair<<<(2 * F0 * HID + T - 1) / T, T, 0, stream>>>(W0s, W0n, p0, F0, HID);
    pack_pair<<<(2 * HID * HID + T - 1) / T, T, 0, stream>>>(W1s, W1n, p1, HID, HID);
    pack_pair<<<(2 * HID * HID + T - 1) / T, T, 0, stream>>>(W2s, W2n, p2, HID, HID);
    pack_single<<<(HID * NCLS + T - 1) / T, T, 0, stream>>>(Wfc, pfc, HID, NCLS);

    // degrees -> reciprocal (shared by all 3 layers)
    zero_f32<<<(NN + T - 1) / T, T, 0, stream>>>(deg, NN);
    degree_count<<<(NE + T - 1) / T, T, 0, stream>>>(edst, deg, NE);
    make_invdeg<<<(NN + T - 1) / T, T, 0, stream>>>(deg, NN);

    const dim3 gemmGrid(1, (MT + 1) / 2);            // 2 M-tiles per block

    // ---- layer 0: x[NN,128] -> hA[NN,256] ----
    zero_f32<<<((NN * F0) + T - 1) / T, T, 0, stream>>>(agg, NN * F0);
    scatter_edges<<<(NE * (F0 / 4)) / T, T, 0, stream>>>(x, esrc, edst, agg, NE, F0, 5);
    sage_gemm_wmma<<<gemmGrid, 128, 0, stream>>>(x, agg, deg, p0, b0, hA, F0, HID, MT);

    // ---- layer 1: hA -> hB ----
    zero_f32<<<((NN * HID) + T - 1) / T, T, 0, stream>>>(agg, NN * HID);
    scatter_edges<<<(NE * (HID / 4)) / T, T, 0, stream>>>(hA, esrc, edst, agg, NE, HID, 6);
    sage_gemm_wmma<<<gemmGrid, 128, 0, stream>>>(hA, agg, deg, p1, b1, hB, HID, HID, MT);

    // ---- layer 2: hB -> hA (buffer reuse) ----
    zero_f32<<<((NN * HID) + T - 1) / T, T, 0, stream>>>(agg, NN * HID);
    scatter_edges<<<(NE * (HID / 4)) / T, T, 0, stream>>>(hB, esrc, edst, agg, NE, HID, 6);
    sage_gemm_wmma<<<gemmGrid, 128, 0, stream>>>(hB, agg, deg, p2, b2, hA, HID, HID, MT);

    // ---- FC: hA[NN,256] @ Wfc[256,64] + bfc -> out[NN,64] ----
    fc_gemm_wmma<<<dim3(1, (MT + 3) / 4), 128, 0, stream>>>(hA, pfc, bfc, out, HID, NCLS, MT);
}